// GCN_19756849561887
// MI455X (gfx1250) — compile-verified
//
#include <hip/hip_runtime.h>
#include <hip/hip_bf16.h>

typedef float v2f __attribute__((ext_vector_type(2)));
typedef float v8f __attribute__((ext_vector_type(8)));

#define N_NODES 50000
#define N_EDGES 800000

// ---------------------------------------------------------------------------
// degree / normalization
// ---------------------------------------------------------------------------
__global__ void k_init_deg(float* __restrict__ deg, int n) {
    int i = blockIdx.x * blockDim.x + threadIdx.x;
    if (i < n) deg[i] = 1.0f;   // self-loop contribution
}

__global__ void k_count_deg(const int* __restrict__ col, float* __restrict__ deg, int e) {
    int i = blockIdx.x * blockDim.x + threadIdx.x;
    if (i < e) atomicAdd(&deg[col[i]], 1.0f);
}

__global__ void k_rsqrt(float* __restrict__ deg, int n) {
    int i = blockIdx.x * blockDim.x + threadIdx.x;
    if (i < n) deg[i] = rsqrtf(deg[i]);   // deg >= 1 always
}

__global__ void k_zero(float* __restrict__ p, int n) {
    int i = blockIdx.x * blockDim.x + threadIdx.x;
    if (i < n) p[i] = 0.0f;
}

// ---------------------------------------------------------------------------
// fp32 WMMA GEMM:  C[M x NOUT] = A[M x K] * W[K x NOUT]
// one wave -> one 16x16 output tile; blockDim = 32 * (NOUT/16); grid.x = M/16
// ---------------------------------------------------------------------------
template<int K, int NOUT>
__global__ void k_gemm_wmma(const float* __restrict__ A,
                            const float* __restrict__ W,
                            float* __restrict__ C) {
    const int lane   = threadIdx.x & 31;
    const int wave   = threadIdx.x >> 5;          // 0 .. NOUT/16-1  (tile_n)
    const int tile_m = blockIdx.x;
    const int mrow   = tile_m * 16 + (lane & 15); // A row for this lane
    const int ncol   = wave * 16 + (lane & 15);   // B/C col for this lane
    const int kh     = (lane >> 4) * 2;           // K sub-offset: 0 or 2

    const float* __restrict__ arow = A + (size_t)mrow * K;

    v8f acc = {};
#pragma unroll 8
    for (int k = 0; k < K; k += 4) {
        v2f a, b;
        // A 16x4 fragment: VGPR j holds K = kh + j
        a[0] = arow[k + kh + 0];
        a[1] = arow[k + kh + 1];
        // B 4x16 fragment: VGPR j holds row K = kh + j, col = lane%16
        b[0] = W[(size_t)(k + kh + 0) * NOUT + ncol];
        b[1] = W[(size_t)(k + kh + 1) * NOUT + ncol];
        acc = __builtin_amdgcn_wmma_f32_16x16x4_f32(
            /*neg_a=*/false, a, /*neg_b=*/false, b,
            /*c_mod=*/(short)0, acc, /*reuse_a=*/false, /*reuse_b=*/false);
    }

    // C/D layout: VGPR r -> row (r + 8*(lane/16)), col = lane%16
    const int crow0 = tile_m * 16 + (lane >> 4) * 8;
#pragma unroll
    for (int r = 0; r < 8; ++r)
        C[(size_t)(crow0 + r) * NOUT + ncol] = acc[r];
}

// ---------------------------------------------------------------------------
// edge scatter:  agg[col] += h[row] * dis[row]*dis[col]
// one wave per edge; D/32 floats per lane; f32 global atomics (L2-resident)
// ---------------------------------------------------------------------------
template<int D>
__global__ void k_edge_scatter(const int* __restrict__ row,
                               const int* __restrict__ col,
                               const float* __restrict__ dis,
                               const float* __restrict__ h,
                               float* __restrict__ agg, int e) {
    const int wavesPerBlock = blockDim.x >> 5;
    int edge = blockIdx.x * wavesPerBlock + (threadIdx.x >> 5);
    if (edge >= e) return;
    const int lane = threadIdx.x & 31;
    const int r = row[edge];
    const int c = col[edge];
    const float norm = dis[r] * dis[c];

    constexpr int F = D / 32;                       // floats per lane
    const float* __restrict__ src = h   + (size_t)r * D + lane * F;
    float*       __restrict__ dst = agg + (size_t)c * D + lane * F;
#pragma unroll
    for (int i = 0; i < F; ++i)
        atomicAdd(&dst[i], src[i] * norm);
}

// ---------------------------------------------------------------------------
// epilogues: add self-loop term h*dis^2 + bias (+ optional ReLU)
// ---------------------------------------------------------------------------
template<int D, bool RELU>
__global__ void k_epilogue(const float* __restrict__ agg,
                           const float* __restrict__ h,
                           const float* __restrict__ dis,
                           const float* __restrict__ bias,
                           float* __restrict__ out, int n) {
    int idx = blockIdx.x * blockDim.x + threadIdx.x;
    if (idx >= n * D) return;
    int i = idx / D;
    int f = idx - i * D;
    float d = dis[i];
    float v = agg[idx] + h[idx] * d * d + bias[f];
    if (RELU) v = v > 0.0f ? v : 0.0f;
    out[idx] = v;
}

// ---------------------------------------------------------------------------
// launch
// ---------------------------------------------------------------------------
extern "C" void kernel_launch(void* const* d_in, const int* in_sizes, int n_in,
                              void* d_out, int out_size, void* d_ws, size_t ws_size,
                              hipStream_t stream) {
    const float* x  = (const float*)d_in[0];
    const int*   ei = (const int*)  d_in[1];   // [2, E] flat
    const float* W1 = (const float*)d_in[2];
    const float* b1 = (const float*)d_in[3];
    const float* W2 = (const float*)d_in[4];
    const float* b2 = (const float*)d_in[5];

    const int N = N_NODES;
    const int E = N_EDGES;
    const int* row = ei;        // edge_index[0]
    const int* col = ei + E;    // edge_index[1]

    // workspace layout (floats)
    float* ws   = (float*)d_ws;
    float* dis  = ws;                        // N
    float* bufA = ws + N;                    // N*128  (h, later h2|agg2)
    float* bufB = bufA + (size_t)N * 128;    // N*128  (agg1 -> h1 in place)
    float* h    = bufA;
    float* agg1 = bufB;
    float* h1   = bufB;
    float* h2   = bufA;                      // N*64
    float* agg2 = bufA + (size_t)N * 64;     // N*64

    const int T = 256;
    // 1) degree -> dis = deg^{-1/2}
    k_init_deg <<<(N + T - 1) / T, T, 0, stream>>>(dis, N);
    k_count_deg<<<(E + T - 1) / T, T, 0, stream>>>(col, dis, E);
    k_rsqrt    <<<(N + T - 1) / T, T, 0, stream>>>(dis, N);

    // 2) h = x @ W1   (WMMA fp32, 16x16 tile per wave, 8 waves/block)
    k_gemm_wmma<128, 128><<<N / 16, 256, 0, stream>>>(x, W1, h);

    // 3) layer-1 aggregation
    k_zero<<<((N * 128) + T - 1) / T, T, 0, stream>>>(agg1, N * 128);
    k_edge_scatter<128><<<(E + 7) / 8, 256, 0, stream>>>(row, col, dis, h, agg1, E);

    // 4) h1 = relu(agg1 + h*dis^2 + b1)   (in place in agg1)
    k_epilogue<128, true><<<((N * 128) + T - 1) / T, T, 0, stream>>>(agg1, h, dis, b1, h1, N);

    // 5) h2 = h1 @ W2   (WMMA fp32, 4 waves/block)
    k_gemm_wmma<128, 64><<<N / 16, 128, 0, stream>>>(h1, W2, h2);

    // 6) layer-2 aggregation
    k_zero<<<((N * 64) + T - 1) / T, T, 0, stream>>>(agg2, N * 64);
    k_edge_scatter<64><<<(E + 7) / 8, 256, 0, stream>>>(row, col, dis, h2, agg2, E);

    // 7) out = agg2 + h2*dis^2 + b2  -> d_out (plain stores, deterministic)
    k_epilogue<64, false><<<((N * 64) + T - 1) / T, T, 0, stream>>>(agg2, h2, dis, b2,
                                                                    (float*)d_out, N);
}